// DirectMPIGO_29420525977640
// MI455X (gfx1250) — compile-verified
//
#include <hip/hip_runtime.h>
#include <hip/hip_bf16.h>

#define WAVE 32

// ---------------------------------------------------------------------------
// Kernel 1: per-ray segment boundaries via binary search on the sorted ray_id.
// starts[r] = lower_bound(ray_id, r); starts[n_rays] = M. One thread per entry.
// ray_id (32 MB) is fully L2-resident on MI455X (192 MB L2), so the ~23 probe
// loads per thread are cheap.
// ---------------------------------------------------------------------------
__global__ void __launch_bounds__(256) dvgo_find_starts(
    const int* __restrict__ ray_id, int m, int n_rays, int* __restrict__ starts) {
  const int r = blockIdx.x * blockDim.x + threadIdx.x;
  if (r > n_rays) return;
  int lo = 0, hi = m;
  while (lo < hi) {
    const int mid = (lo + hi) >> 1;
    if (ray_id[mid] < r) lo = mid + 1; else hi = mid;
  }
  starts[r] = lo;
}

// ---------------------------------------------------------------------------
// Kernel 2: one wave32 per ray. Tiles of 32 samples:
//   lp    = -softplus(d + shift) * interval          (= log(1 - alpha))
//   scan  = wave-inclusive prefix sum of lp (5x shfl_up, ds-permute path)
//   logT  = carry + (scan - lp)                       (exclusive, log space)
//   w     = exp(logT) * (-expm1(lp))
// carry accumulates lane31's inclusive total across tiles; lane 0 finally
// writes alphainv_last = exp(carry) = prod(1-alpha).
// Invalid (tail) lanes contribute lp = 0, so the scan/carry stay exact.
// ---------------------------------------------------------------------------
__global__ void __launch_bounds__(256) dvgo_render_weights(
    const float* __restrict__ density,
    const float* __restrict__ shift_p,
    const float* __restrict__ interval_p,
    const int*   __restrict__ starts,
    float* __restrict__ weights,
    float* __restrict__ alphainv_last,
    int n_rays) {
  const int gtid = blockIdx.x * blockDim.x + threadIdx.x;
  const int ray  = gtid >> 5;           // one wave32 per ray
  const int lane = gtid & (WAVE - 1);
  if (ray >= n_rays) return;

  const float shift    = *shift_p;      // uniform scalar loads
  const float interval = *interval_p;
  const int s = starts[ray];
  const int e = starts[ray + 1];

  float carry = 0.0f;                   // running sum of lp = log T
  for (int base = s; base < e; base += WAVE) {
    const int idx = base + lane;
    const bool valid = idx < e;

    // gfx1250 prefetch of the next tile (global_prefetch_b8)
    if (base + WAVE < e) __builtin_prefetch(&density[idx + WAVE], 0, 0);

    // Stable softplus, identical formulation to jax.nn.softplus.
    const float x  = (valid ? density[idx] : 0.0f) + shift;
    const float sp = fmaxf(x, 0.0f) + log1pf(expf(-fabsf(x)));
    const float lp = valid ? (-sp * interval) : 0.0f;

    // Wave32 inclusive Hillis-Steele scan (5 steps).
    float scan = lp;
#pragma unroll
    for (int off = 1; off < WAVE; off <<= 1) {
      const float up = __shfl_up(scan, off, WAVE);
      if (lane >= off) scan += up;
    }

    const float logT = carry + (scan - lp);          // exclusive within ray
    if (valid) {
      const float alpha = -expm1f(lp);               // 1 - exp(lp)
      weights[idx] = expf(logT) * alpha;
    }
    carry += __shfl(scan, WAVE - 1, WAVE);           // broadcast tile total
  }

  if (lane == 0) alphainv_last[ray] = expf(carry);
}

// ---------------------------------------------------------------------------
// Launch. Inputs (setup_inputs order): density[M] f32, shift f32[1],
// interval f32[1], ray_id[M] i32, n_rays i32[1]. Output: weights[M] ++
// alphainv_last[N] (f32). n_rays derived on host as out_size - M so no device
// reads are needed (graph-capture safe). d_ws: (N+1) ints of scratch.
// ---------------------------------------------------------------------------
extern "C" void kernel_launch(void* const* d_in, const int* in_sizes, int n_in,
                              void* d_out, int out_size, void* d_ws, size_t ws_size,
                              hipStream_t stream) {
  const float* density  = (const float*)d_in[0];
  const float* shift    = (const float*)d_in[1];
  const float* interval = (const float*)d_in[2];
  const int*   ray_id   = (const int*)d_in[3];

  const int m      = in_sizes[0];
  const int n_rays = out_size - m;

  float* weights       = (float*)d_out;
  float* alphainv_last = weights + m;
  int*   starts        = (int*)d_ws;   // (n_rays + 1) ints

  {
    const int total  = n_rays + 1;
    const int blocks = (total + 255) / 256;
    dvgo_find_starts<<<blocks, 256, 0, stream>>>(ray_id, m, n_rays, starts);
  }
  {
    const int waves_per_block = 256 / WAVE;            // 8 waves / block
    const int blocks = (n_rays + waves_per_block - 1) / waves_per_block;
    dvgo_render_weights<<<blocks, 256, 0, stream>>>(
        density, shift, interval, starts, weights, alphainv_last, n_rays);
  }
}